// Encoder_9010841387203
// MI455X (gfx1250) — compile-verified
//
#include <hip/hip_runtime.h>

// CDNA5 / gfx1250 fused encoder:
//   circular-H 5x5 conv (center tap zeroed, C=1, O=20)  -> WMMA f32 16x16x4 (im2col from LDS)
//   [W21*y ; (W22*y)*(W23*y)]                            -> WMMA f32 16x16x4
//   W3 (40->1)                                           -> VALU reduce + shfl_xor(16)
// Fully fused: one HBM read of xin, one HBM write of out; 84MB intermediate lives in LDS tiles.

typedef __attribute__((ext_vector_type(2))) float v2f;
typedef __attribute__((ext_vector_type(8))) float v8f;

#define Hh 128
#define Ww 128
#define CH 20
#define ROWS 4
#define TILE_PIX (ROWS * Ww)   // 512 pixels per block
#define XS_ROWS (ROWS + 4)     // 8 rows incl. circular halo
#define XS_COLS (Ww + 4)       // 132 cols incl. zero halo
#define NTILES (TILE_PIX / 16) // 32 N-tiles of 16 pixels

static __device__ __forceinline__ v8f wmma4f32(v2f a, v2f b, v8f c) {
  // D = A(16x4,f32) * B(4x16,f32) + C(16x16,f32)
  return __builtin_amdgcn_wmma_f32_16x16x4_f32(false, a, false, b, (short)0, c,
                                               false, false);
}

__global__ __launch_bounds__(256) void encoder_fused(
    const float* __restrict__ xin, const float* __restrict__ w1,
    const float* __restrict__ w21, const float* __restrict__ w22,
    const float* __restrict__ w23, const float* __restrict__ w3,
    float* __restrict__ out) {
  __shared__ float xs[XS_ROWS * XS_COLS];   // haloed input strip
  __shared__ float ys[CH * TILE_PIX];       // conv output, [channel][pixel]

  const int b     = blockIdx.x;
  const int strip = blockIdx.y;
  const int r0    = strip * ROWS;
  const int tid   = threadIdx.x;
  const int lane  = tid & 31;
  const int wave  = tid >> 5;
  const int ln16  = lane & 15;
  const int hi    = lane >> 4;  // 0 = lanes 0-15, 1 = lanes 16-31

  // ---- Phase 1: stage input strip in LDS (circular H halo, zero W halo) ----
  const float* xb = xin + b * (Hh * Ww);
  for (int idx = tid; idx < XS_ROWS * XS_COLS; idx += 256) {
    const int i = idx / XS_COLS;
    const int j = idx - i * XS_COLS;
    float v = 0.0f;
    if (j >= 2 && j < Ww + 2) {
      const int srow = (r0 - 2 + i) & (Hh - 1);  // circular wrap in H
      v = xb[srow * Ww + (j - 2)];
    }
    xs[idx] = v;
  }
  __syncthreads();

  // ---- Wave-invariant A fragments ----
  // Conv weights as im2col matrix: A[m][k], K = 25 taps padded to 28, tap 12 zeroed.
  v2f Ac[2][7];
#pragma unroll
  for (int mt = 0; mt < 2; ++mt) {
    const int m = mt * 16 + ln16;
#pragma unroll
    for (int kc = 0; kc < 7; ++kc) {
      const int ka = kc * 4 + 2 * hi;  // A layout: V0 -> k0 + 2*hi
      const int kb = ka + 1;           //           V1 -> k0 + 1 + 2*hi
      const float a0 = (m < CH && ka < 25 && ka != 12) ? w1[m * 25 + ka] : 0.0f;
      const float a1 = (m < CH && kb < 25 && kb != 12) ? w1[m * 25 + kb] : 0.0f;
      v2f a = {a0, a1};
      Ac[mt][kc] = a;
    }
  }
  // Pointwise 20x20 weights, K = 20 = 5 chunks of 4 exactly.
  v2f Ap[3][2][5];
  const float* Wp[3] = {w21, w22, w23};
#pragma unroll
  for (int wsel = 0; wsel < 3; ++wsel) {
#pragma unroll
    for (int mt = 0; mt < 2; ++mt) {
      const int m = mt * 16 + ln16;
#pragma unroll
      for (int kc = 0; kc < 5; ++kc) {
        const int ka = kc * 4 + 2 * hi;
        const float a0 = (m < CH) ? Wp[wsel][m * CH + ka] : 0.0f;
        const float a1 = (m < CH) ? Wp[wsel][m * CH + ka + 1] : 0.0f;
        v2f a = {a0, a1};
        Ap[wsel][mt][kc] = a;
      }
    }
  }
  // W3 broadcast values for the final 40->1 reduction.
  float w3a[8], w3b[8], w3c[4], w3d[4];
#pragma unroll
  for (int v = 0; v < 8; ++v) {
    w3a[v] = w3[v + 8 * hi];        // channels 0..15   (C-tile 0 rows)
    w3b[v] = w3[20 + v + 8 * hi];   // channels 20..35
  }
#pragma unroll
  for (int v = 0; v < 4; ++v) {
    w3c[v] = w3[16 + v];            // channels 16..19  (C-tile 1, hi==0, v<4)
    w3d[v] = w3[36 + v];            // channels 36..39
  }

  // ---- Phase 2: per 16-pixel N-tile, wave-local conv -> pointwise -> reduce ----
  for (int t = wave; t < NTILES; t += 8) {
    const int n0 = t * 16;
    const int ry = n0 >> 7;                 // row within strip (tiles never span rows)
    const int cx = (n0 & (Ww - 1)) + ln16;  // this lane's output column

    // 5x5 conv via implicit im2col: B[k][n] gathered straight from xs.
    v8f c0 = {0, 0, 0, 0, 0, 0, 0, 0};
    v8f c1 = {0, 0, 0, 0, 0, 0, 0, 0};
#pragma unroll
    for (int kc = 0; kc < 7; ++kc) {
      const int ta = kc * 4 + 2 * hi;  // B layout: V0 -> k0 (lo half) / k0+2 (hi half)
      const int tb = ta + 1;           //           V1 -> k0+1 / k0+3
      float b0 = 0.0f, b1 = 0.0f;
      if (ta < 25) {
        const int dy = ta / 5, dx = ta - dy * 5;
        b0 = xs[(ry + dy) * XS_COLS + cx + dx];
      }
      if (tb < 25) {
        const int dy = tb / 5, dx = tb - dy * 5;
        b1 = xs[(ry + dy) * XS_COLS + cx + dx];
      }
      v2f bf = {b0, b1};
      c0 = wmma4f32(Ac[0][kc], bf, c0);  // channels 0..15
      c1 = wmma4f32(Ac[1][kc], bf, c1);  // channels 16..19 (rows 20..31 garbage, never read)
    }

    // Spill conv tile to LDS in [channel][pixel] form (wave-local, in-order DS).
#pragma unroll
    for (int v = 0; v < 8; ++v)
      ys[(v + 8 * hi) * TILE_PIX + n0 + ln16] = c0[v];
#pragma unroll
    for (int v = 0; v < 4; ++v)
      if (hi == 0) ys[(16 + v) * TILE_PIX + n0 + ln16] = c1[v];

    // Three 20x20 pointwise matmuls sharing B fragments.
    v8f d[3][2];
#pragma unroll
    for (int wsel = 0; wsel < 3; ++wsel)
#pragma unroll
      for (int mt = 0; mt < 2; ++mt) {
        v8f z = {0, 0, 0, 0, 0, 0, 0, 0};
        d[wsel][mt] = z;
      }
#pragma unroll
    for (int kc = 0; kc < 5; ++kc) {
      const int c = kc * 4 + 2 * hi;
      const float b0 = ys[c * TILE_PIX + n0 + ln16];
      const float b1 = ys[(c + 1) * TILE_PIX + n0 + ln16];
      v2f bf = {b0, b1};
#pragma unroll
      for (int wsel = 0; wsel < 3; ++wsel) {
        d[wsel][0] = wmma4f32(Ap[wsel][0][kc], bf, d[wsel][0]);
        d[wsel][1] = wmma4f32(Ap[wsel][1][kc], bf, d[wsel][1]);
      }
    }

    // Final 40 -> 1: out = sum_c w3[c]*t1[c] + w3[20+c]*(t2[c]*t3[c]).
    float part = 0.0f;
#pragma unroll
    for (int v = 0; v < 8; ++v)
      part += w3a[v] * d[0][0][v] + w3b[v] * (d[1][0][v] * d[2][0][v]);
    if (hi == 0) {
#pragma unroll
      for (int v = 0; v < 4; ++v)
        part += w3c[v] * d[0][1][v] + w3d[v] * (d[1][1][v] * d[2][1][v]);
    }
    // rows of a column are split across lane halves: fold lane n with lane n+16
    part += __shfl_xor(part, 16, 32);
    if (hi == 0) out[b * (Hh * Ww) + (r0 + ry) * Ww + cx] = part;
  }
}

extern "C" void kernel_launch(void* const* d_in, const int* in_sizes, int n_in,
                              void* d_out, int out_size, void* d_ws,
                              size_t ws_size, hipStream_t stream) {
  const float* xin = (const float*)d_in[0];
  const float* w1  = (const float*)d_in[1];
  const float* w21 = (const float*)d_in[2];
  const float* w22 = (const float*)d_in[3];
  const float* w23 = (const float*)d_in[4];
  const float* w3  = (const float*)d_in[5];
  float* out = (float*)d_out;

  const int B = in_sizes[0] / (Hh * Ww);   // 64
  dim3 grid(B, Hh / ROWS);                 // (64, 32)
  encoder_fused<<<grid, 256, 0, stream>>>(xin, w1, w21, w22, w23, w3, out);
}